// ModalConditionedRPE_50534585205058
// MI455X (gfx1250) — compile-verified
//
#include <hip/hip_runtime.h>

#define H 40
#define W 40
#define NPIX (H * W)   // 1600
#define HID 64
#define TILE 16
#define LDS_S (HID + 1)   // pad to avoid 64-bank stride conflicts

typedef __attribute__((ext_vector_type(2))) float v2f;
typedef __attribute__((ext_vector_type(8))) float v8f;

// 1D taps for jax.image.resize(linear, antialias=True), 80 -> 40.
// Interior: input coord 2o+0.5, triangle kernel scaled by 2 -> [1/8,3/8,3/8,1/8].
// Edges: out-of-range taps dropped, weights renormalized (sum 1.75 -> /7).
__device__ __forceinline__ void taps40(int o, int* idx, float* wt) {
    if (o == 0) {
        idx[0] = 0; idx[1] = 1; idx[2] = 2; idx[3] = 2;
        wt[0] = 3.f / 7.f; wt[1] = 3.f / 7.f; wt[2] = 1.f / 7.f; wt[3] = 0.f;
    } else if (o == W - 1) {
        idx[0] = 77; idx[1] = 78; idx[2] = 79; idx[3] = 79;
        wt[0] = 1.f / 7.f; wt[1] = 3.f / 7.f; wt[2] = 3.f / 7.f; wt[3] = 0.f;
    } else {
        idx[0] = 2 * o - 1; idx[1] = 2 * o; idx[2] = 2 * o + 1; idx[3] = 2 * o + 2;
        wt[0] = 0.125f; wt[1] = 0.375f; wt[2] = 0.375f; wt[3] = 0.125f;
    }
}

// s_mean[1600] = mean over 4 batches of antialiased 80x80 -> 40x40 resize.
// base[64]    = mod_vec @ W_mod + b1.
__global__ void prep_kernel(const float* __restrict__ smap,      // (4,1,80,80)
                            const float* __restrict__ mod_embed, // (3,16)
                            const float* __restrict__ W1,        // (36,64)
                            const float* __restrict__ b1,        // (64,)
                            const int* __restrict__ qmod_p,
                            const int* __restrict__ kmod_p,
                            float* __restrict__ s_mean,
                            float* __restrict__ base) {
    int tid = blockIdx.x * blockDim.x + threadIdx.x;
    if (tid < NPIX) {
        int oy = tid / W, ox = tid % W;
        int iy[4], ix[4];
        float wy[4], wx[4];
        taps40(oy, iy, wy);
        taps40(ox, ix, wx);
        float acc = 0.f;
        for (int bb = 0; bb < 4; ++bb) {
            const float* m = smap + bb * 6400;
            float v = 0.f;
            #pragma unroll
            for (int a = 0; a < 4; ++a) {
                float row = 0.f;
                #pragma unroll
                for (int b = 0; b < 4; ++b) row += wx[b] * m[iy[a] * 80 + ix[b]];
                v += wy[a] * row;
            }
            acc += v;
        }
        s_mean[tid] = acc * 0.25f;
    } else if (tid < NPIX + HID) {
        int hh = tid - NPIX;
        int q = qmod_p[0], k = kmod_p[0];
        float acc = b1[hh];
        for (int c = 0; c < 16; ++c) acc = fmaf(mod_embed[q * 16 + c], W1[(2 + c) * HID + hh], acc);
        for (int c = 0; c < 16; ++c) acc = fmaf(mod_embed[k * 16 + c], W1[(18 + c) * HID + hh], acc);
        base[hh] = acc;
    }
}

// One block = one 16x16 output tile.
// Wave 0: A-tile = [x,y,s,1](16x4) x [W1_0;W1_1;W1_34;base](4x64) via 4x v_wmma_f32_16x16x4_f32.
// Wave 1: B-tile = [-x,-y,s,0]    x [W1_0;W1_1;W1_35;  * ].
// All waves: out[i,j] = b2 + sum_h relu(A[il,h]+B[jl,h]) * W2[h].
__global__ __launch_bounds__(256) void rpe_kernel(const float* __restrict__ W1,
                                                  const float* __restrict__ W2,
                                                  const float* __restrict__ b2,
                                                  const float* __restrict__ s_mean,
                                                  const float* __restrict__ base,
                                                  float* __restrict__ out) {
    __shared__ float As[TILE][LDS_S];
    __shared__ float Bs[TILE][LDS_S];
    __shared__ float w2s[HID];

    const int tid  = threadIdx.x;
    const int wave = tid >> 5;
    const int lane = tid & 31;
    const int i0 = blockIdx.y * TILE;
    const int j0 = blockIdx.x * TILE;

    if (tid < HID) w2s[tid] = W2[tid];   // waves 0,1 fully covered: no intra-wave divergence

    if (wave < 2) {
        const int  m   = lane & 15;
        const bool hiK = (lane >= 16);            // lanes 16-31 hold K=2,3
        const int  row = (wave == 0 ? i0 : j0) + m;
        const int  ry  = row / W, rx = row % W;
        const float x = -0.5f + (float)rx * (1.0f / 39.0f);
        const float y = -0.5f + (float)ry * (1.0f / 39.0f);
        const float s = s_mean[row];

        // features for this wave's matrix
        const float f0 = (wave == 0) ? x : -x;
        const float f1 = (wave == 0) ? y : -y;
        const float f2 = s;
        const float f3 = (wave == 0) ? 1.0f : 0.0f;

        v2f a;
        a.x = hiK ? f2 : f0;  // K = hiK?2:0
        a.y = hiK ? f3 : f1;  // K = hiK?3:1

        const int srow = (wave == 0) ? 34 : 35;   // W_si / W_sj row of W1
        const int n = lane & 15;                  // output column within h-tile

        #pragma unroll
        for (int t = 0; t < 4; ++t) {
            const int hc = t * 16 + n;
            const float r0 = W1[0 * HID + hc];
            const float r1 = W1[1 * HID + hc];
            const float r2 = W1[srow * HID + hc];
            const float r3 = base[hc];            // x0 feature for wave 1 -> ignored
            v2f bop;
            bop.x = hiK ? r2 : r0;
            bop.y = hiK ? r3 : r1;
            v8f c = {};
            v8f d = __builtin_amdgcn_wmma_f32_16x16x4_f32(false, a, false, bop,
                                                          (short)0, c, false, false);
            float* dst = (wave == 0) ? &As[0][0] : &Bs[0][0];
            #pragma unroll
            for (int r = 0; r < 8; ++r) {
                const int mr = r + (hiK ? 8 : 0); // D: VGPR r holds rows r and r+8
                dst[mr * LDS_S + hc] = d[r];
            }
        }
    }
    __syncthreads();

    const int jl = tid & 15;
    const int il = tid >> 4;
    float acc = b2[0];
    #pragma unroll 16
    for (int h = 0; h < HID; ++h) {
        float v = As[il][h] + Bs[jl][h];
        v = fmaxf(v, 0.0f);
        acc = fmaf(v, w2s[h], acc);
    }
    out[(size_t)(i0 + il) * NPIX + (j0 + jl)] = acc;
}

extern "C" void kernel_launch(void* const* d_in, const int* in_sizes, int n_in,
                              void* d_out, int out_size, void* d_ws, size_t ws_size,
                              hipStream_t stream) {
    // setup_inputs order: h, w, q_mod, k_mod, structure_map, mod_embed, W1, b1, W2, b2
    const int*   qmod = (const int*)d_in[2];
    const int*   kmod = (const int*)d_in[3];
    const float* smap = (const float*)d_in[4];
    const float* mode = (const float*)d_in[5];
    const float* W1   = (const float*)d_in[6];
    const float* b1   = (const float*)d_in[7];
    const float* W2   = (const float*)d_in[8];
    const float* b2   = (const float*)d_in[9];

    float* ws     = (float*)d_ws;
    float* s_mean = ws;            // 1600 floats
    float* base   = ws + NPIX;     // 64 floats

    prep_kernel<<<(NPIX + HID + 255) / 256, 256, 0, stream>>>(smap, mode, W1, b1,
                                                              qmod, kmod, s_mean, base);

    dim3 grid(NPIX / TILE, NPIX / TILE);   // 100 x 100 tiles
    rpe_kernel<<<grid, 256, 0, stream>>>(W1, W2, b2, s_mean, base, (float*)d_out);
}